// JPEGCompression_52785148068110
// MI455X (gfx1250) — compile-verified
//
#include <hip/hip_runtime.h>
#include <hip/hip_bf16.h>

// JPEG compression round-trip, fully fused single pass.
// One wave32 handles one 16x16 pixel tile (= four 8x8 JPEG blocks) for all
// 3 channels. DCT via V_WMMA_F32_16X16X4_F32 with the block-diagonal basis
// Dblk = diag(M, M):  Z = Dblk * X * Dblk^T  computes all four 8x8 DCTs.
// Each matmul stage is "left-multiply then transpose" (transpose bounced
// through padded LDS, pitch 21 -> conflict-free), so the constant basis is
// always the A operand and only B needs re-layout.

typedef float v2f __attribute__((ext_vector_type(2)));
typedef float v8f __attribute__((ext_vector_type(8)));

#define IMG_W 512
#define CH_STRIDE (IMG_W * IMG_W)
#define LDS_PITCH 21 // 16 + 5 pad: conflict-free for both scatter and gather

// Full 16x16x16 product as 4 chained K=4 WMMAs. A, B are per-lane slices:
//   A[s] = { Dblk[r][4s+2h], Dblk[r][4s+2h+1] }       (ISA 16x4 A layout)
//   B[s] = { T[4s+2h][r],    T[4s+1+2h][r] }          (ISA 4x16 B layout)
__device__ __forceinline__ v8f wmma_mul16(const v2f A[4], const v2f B[4]) {
  v8f acc = {0.f, 0.f, 0.f, 0.f, 0.f, 0.f, 0.f, 0.f};
#pragma unroll
  for (int s = 0; s < 4; ++s) {
    acc = __builtin_amdgcn_wmma_f32_16x16x4_f32(
        false, A[s], false, B[s], (short)0, acc, false, false);
  }
  return acc;
}

// D-layout result y (VGPR v holds Y[v + 8h][r]) -> LDS holds Y^T:
// ldsT[j*PITCH + i] = Y[i][j]
__device__ __forceinline__ void xpose_store(float* lds, v8f y, int h, int r) {
#pragma unroll
  for (int v = 0; v < 8; ++v) lds[r * LDS_PITCH + v + 8 * h] = y[v];
}

// Read transposed matrix T back in B-operand layout.
__device__ __forceinline__ void xpose_load(const float* lds, v2f b[4], int h, int r) {
#pragma unroll
  for (int s = 0; s < 4; ++s) {
    b[s].x = lds[(4 * s + 2 * h) * LDS_PITCH + r];
    b[s].y = lds[(4 * s + 1 + 2 * h) * LDS_PITCH + r];
  }
}

// tile: 16x16 channel data (B layout). Returns reconstructed pixels, with
// out[v] = R[2v + h][r]  (store-friendly layout).
__device__ __forceinline__ v8f compress_tile(const v2f tile[4],
                                             const v2f a_fwd[4],
                                             const v2f a_inv[4],
                                             const float qrow[8],
                                             float* lds, int h, int r) {
  v2f t[4];
  // Stage 1: Y1 = Dblk * X               -> LDS: X^T * Dblk^T
  v8f y = wmma_mul16(a_fwd, tile);
  xpose_store(lds, y, h, r);
  xpose_load(lds, t, h, r);
  // Stage 2: Y2 = Dblk * X^T * Dblk^T = Z^T  (D layout)
  y = wmma_mul16(a_fwd, t);
  // Quantize Z^T: element (i = v+8h, j = r) is coeff (u = r&7, v_c = v)
  // -> Q[r&7][v] = qrow[v]  (contiguous row, preloaded)
#pragma unroll
  for (int v = 0; v < 8; ++v) {
    float q = qrow[v];
    y[v] = rintf(y[v] / q) * q;
  }
  xpose_store(lds, y, h, r); // LDS: Zq
  xpose_load(lds, t, h, r);
  // Stage 3: Y3 = Dblk^T * Zq            -> LDS: Zq^T * Dblk
  y = wmma_mul16(a_inv, t);
  xpose_store(lds, y, h, r);
  xpose_load(lds, t, h, r);
  // Stage 4: Y4 = Dblk^T * Zq^T * Dblk = R^T -> LDS: R
  y = wmma_mul16(a_inv, t);
  xpose_store(lds, y, h, r);
  v8f out;
#pragma unroll
  for (int v = 0; v < 8; ++v) out[v] = lds[(2 * v + h) * LDS_PITCH + r];
  return out;
}

__global__ __launch_bounds__(256) void JPEGCompression_gfx1250_kernel(
    const float* __restrict__ img, const float* __restrict__ rgb2yuv,
    const float* __restrict__ yuv2rgb, const float* __restrict__ dct,
    const float* __restrict__ qlum, const float* __restrict__ qchrom,
    float* __restrict__ out) {
  __shared__ float lds_all[8][16 * LDS_PITCH];
  const int lane = threadIdx.x & 31;
  const int warp = threadIdx.x >> 5;
  float* lds = lds_all[warp];
  const int h = lane >> 4;  // lane half
  const int r = lane & 15;  // row (A) / column (B,C,D) index

  const int wid = blockIdx.x * 8 + warp; // one wave == one (batch, tile)
  const int batch = wid >> 10;           // 1024 tiles per 512x512 image
  const int tile = wid & 1023;
  const int row0 = (tile >> 5) * 16;
  const int col0 = (tile & 31) * 16;

  // ---- Basis operands from dct_coeff: M[u][x] = C[u,0,x,0] * 2*sqrt(2) ----
  const float SC = 2.82842712474619009760f; // 2*sqrt(2); M[0][0] = 1/(2*sqrt(2))
  const int uq = r & 7;
  v2f a_fwd[4], a_inv[4];
#pragma unroll
  for (int s = 0; s < 4; ++s) {
    const int c0 = 4 * s + 2 * h, c1 = c0 + 1;
    const bool nz = (r < 8) == (c0 < 8); // block-diagonal mask
    const int x0 = c0 & 7, x1 = c1 & 7;
    const float m00 = dct[uq * 512 + x0 * 8] * SC; // M[uq][x0]
    const float m01 = dct[uq * 512 + x1 * 8] * SC;
    const float m10 = dct[x0 * 512 + uq * 8] * SC; // M[x0][uq]
    const float m11 = dct[x1 * 512 + uq * 8] * SC;
    a_fwd[s].x = nz ? m00 : 0.f;
    a_fwd[s].y = nz ? m01 : 0.f;
    a_inv[s].x = nz ? m10 : 0.f;
    a_inv[s].y = nz ? m11 : 0.f;
  }

  // Quant rows (contiguous) and color matrices.
  float ql[8], qc[8];
#pragma unroll
  for (int v = 0; v < 8; ++v) {
    ql[v] = qlum[uq * 8 + v];
    qc[v] = qchrom[uq * 8 + v];
  }
  float cf[9], cb[9];
#pragma unroll
  for (int i = 0; i < 9; ++i) {
    cf[i] = rgb2yuv[i];
    cb[i] = yuv2rgb[i];
  }

  // ---- Load RGB tile in B layout, convert to (YUV*255 - 128) ----
  const size_t base = (size_t)batch * 3 * CH_STRIDE;
  v2f tY[4], tU[4], tV[4];
#pragma unroll
  for (int s = 0; s < 4; ++s) {
#pragma unroll
    for (int t = 0; t < 2; ++t) {
      const int rowg = row0 + 4 * s + t + 2 * h;
      const size_t off = base + (size_t)rowg * IMG_W + col0 + r;
      const float R = img[off];
      const float G = img[off + CH_STRIDE];
      const float B = img[off + 2 * CH_STRIDE];
      const float Y = cf[0] * R + cf[1] * G + cf[2] * B;
      const float U = cf[3] * R + cf[4] * G + cf[5] * B;
      const float V = cf[6] * R + cf[7] * G + cf[8] * B;
      tY[s][t] = Y * 255.f - 128.f;
      tU[s][t] = U * 255.f - 128.f;
      tV[s][t] = V * 255.f - 128.f;
    }
  }

  // ---- Compress each channel (Y: luma table, U/V: chroma table) ----
  const v8f Ry = compress_tile(tY, a_fwd, a_inv, ql, lds, h, r);
  const v8f Ru = compress_tile(tU, a_fwd, a_inv, qc, lds, h, r);
  const v8f Rv = compress_tile(tV, a_fwd, a_inv, qc, lds, h, r);

  // ---- (x + 128)/255, YUV->RGB, coalesced stores ----
#pragma unroll
  for (int v = 0; v < 8; ++v) {
    const float yc = (Ry[v] + 128.f) * (1.f / 255.f);
    const float uc = (Ru[v] + 128.f) * (1.f / 255.f);
    const float vc = (Rv[v] + 128.f) * (1.f / 255.f);
    const size_t off = base + (size_t)(row0 + 2 * v + h) * IMG_W + col0 + r;
    out[off]                 = cb[0] * yc + cb[1] * uc + cb[2] * vc;
    out[off + CH_STRIDE]     = cb[3] * yc + cb[4] * uc + cb[5] * vc;
    out[off + 2 * CH_STRIDE] = cb[6] * yc + cb[7] * uc + cb[8] * vc;
  }
}

extern "C" void kernel_launch(void* const* d_in, const int* in_sizes, int n_in,
                              void* d_out, int out_size, void* d_ws, size_t ws_size,
                              hipStream_t stream) {
  const float* img     = (const float*)d_in[0]; // (B,3,512,512)
  const float* rgb2yuv = (const float*)d_in[1]; // (3,3)
  const float* yuv2rgb = (const float*)d_in[2]; // (3,3)
  const float* dct     = (const float*)d_in[3]; // (8,8,8,8)
  const float* qlum    = (const float*)d_in[4]; // (8,8)
  const float* qchrom  = (const float*)d_in[5]; // (8,8)
  float* out = (float*)d_out;

  const int batches = in_sizes[0] / (3 * CH_STRIDE); // 16
  const int waves = batches * 1024;                  // one wave per 16x16 tile
  const int blocks = waves / 8;                      // 8 waves (256 thr) per block

  JPEGCompression_gfx1250_kernel<<<blocks, 256, 0, stream>>>(
      img, rgb2yuv, yuv2rgb, dct, qlum, qchrom, out);
}